// Attention_257698038496
// MI455X (gfx1250) — compile-verified
//
#include <hip/hip_runtime.h>
#include <math.h>

typedef __attribute__((ext_vector_type(2))) float v2f;
typedef __attribute__((ext_vector_type(4))) float v4f;
typedef __attribute__((ext_vector_type(8))) float v8f;

#define BB 4
#define HH 16
#define SS 2048
#define DD 64
#define QT 16                  // q rows per block
#define NWAVE 8
#define NTHREADS 256
#define NKT (SS / 16)          // 128 k-tiles of 16
#define SMEM_FLOATS (QT * SS + 16 + NWAVE * 256)
#define SMEM_BYTES (SMEM_FLOATS * 4)

__global__ __launch_bounds__(NTHREADS)
void attn_fused_kernel(const float* __restrict__ q,
                       const float* __restrict__ k,
                       const float* __restrict__ v,
                       const int*   __restrict__ mask,
                       float* __restrict__ out,
                       float* __restrict__ p_attn)
{
    extern __shared__ float smem[];
    float* sc   = smem;                  // [QT][SS] score / exp tile (128 KB)
    float* rinv = smem + QT * SS;        // [16] 1/rowsum
    float* red  = rinv + 16;             // [NWAVE][256] PV partial reduction

    const int tid  = threadIdx.x;
    const int wave = tid >> 5;
    const int lane = tid & 31;
    const int half = lane >> 4;          // 0: lanes 0-15, 1: lanes 16-31
    const int l16  = lane & 15;

    const int blk = blockIdx.x;
    const int bh  = blk >> 7;            // blk / (S/QT)
    const int qt  = blk & (NKT - 1);
    const int b   = bh >> 4;

    const float* qb = q + ((size_t)bh * SS + (size_t)qt * QT) * DD;
    const float* kb = k + (size_t)bh * SS * DD;
    const float* vb = v + (size_t)bh * SS * DD;
    const int*   mb = mask + b * SS;
    float* pb = p_attn + ((size_t)bh * SS + (size_t)qt * QT) * SS;
    float* ob = out + ((size_t)bh * SS + (size_t)qt * QT) * DD;

    // ---- Phase 1: Q-tile A-fragments (fp32 WMMA layout: 16 chunks of K=4) ----
    // lane<16: M=lane, K={c*4+0, c*4+1}; lane>=16: M=lane-16, K={c*4+2, c*4+3}
    v2f afrag[16];
    {
        const float* qrow = qb + l16 * DD + half * 2;
        #pragma unroll
        for (int c = 0; c < 16; ++c)
            afrag[c] = *(const v2f*)(qrow + c * 4);
    }

    // ---- Phase 2: scores = QK^T / sqrt(D), masked, into LDS ----
    // Two independent accumulator chains for XDL ILP.
    for (int kt = wave; kt < NKT; kt += NWAVE) {
        const float* krow = kb + (size_t)(kt * 16 + l16) * DD + half * 2;
        v8f acc0 = {};
        v8f acc1 = {};
        #pragma unroll
        for (int c = 0; c < 16; c += 2) {
            v2f bf0 = *(const v2f*)(krow + c * 4);
            v2f bf1 = *(const v2f*)(krow + c * 4 + 4);
            acc0 = __builtin_amdgcn_wmma_f32_16x16x4_f32(
                false, afrag[c], false, bf0, (short)0, acc0, false, false);
            acc1 = __builtin_amdgcn_wmma_f32_16x16x4_f32(
                false, afrag[c + 1], false, bf1, (short)0, acc1, false, false);
        }
        v8f acc = acc0 + acc1;
        const int mval = mb[kt * 16 + l16];
        // C layout: VGPR j, lane<16 -> (M=j, N=lane); lane>=16 -> (M=j+8, N=lane-16)
        float* scol = sc + (half * 8) * SS + (kt * 16 + l16);
        #pragma unroll
        for (int j = 0; j < 8; ++j) {
            float s = mval ? acc[j] * 0.125f : -1.0e9f;
            scol[j * SS] = s;
        }
    }
    __syncthreads();

    // ---- Phase 3: softmax over full rows held in LDS ----
    {
        const int row = tid >> 4;        // one row per 16-lane group
        const int c0  = tid & 15;
        float* r = sc + row * SS;
        float m = -1.0e30f;
        for (int c = c0; c < SS; c += 16) m = fmaxf(m, r[c]);
        #pragma unroll
        for (int off = 8; off; off >>= 1) m = fmaxf(m, __shfl_xor(m, off, 16));
        float ssum = 0.0f;
        for (int c = c0; c < SS; c += 16) {
            float e = __expf(r[c] - m);
            r[c] = e;                    // keep unnormalized exp in LDS
            ssum += e;
        }
        #pragma unroll
        for (int off = 8; off; off >>= 1) ssum += __shfl_xor(ssum, off, 16);
        if (c0 == 0) rinv[row] = 1.0f / ssum;
    }
    __syncthreads();

    // ---- Phase 4: normalized p_attn -> global, streaming (non-temporal) ----
    // p_attn is written once and never re-read: keep it out of L2 so the
    // K/V working set (64 MB, re-read by 128 blocks per (b,h)) stays resident.
    for (int i4 = tid; i4 < QT * SS / 4; i4 += NTHREADS) {
        const int row = i4 >> 9;         // SS/4 = 512 float4 per row
        v4f pv = *(const v4f*)(sc + i4 * 4);
        pv *= rinv[row];
        __builtin_nontemporal_store(pv, (v4f*)(pb + (size_t)i4 * 4));
    }

    // ---- Phase 5: out = P * V via f32 WMMA (A from LDS, B from global/L2) ----
    const int nt    = wave & 3;          // which 16-col slice of D
    const int khalf = wave >> 2;         // K-range split between wave pairs
    v8f oacc0 = {};
    v8f oacc1 = {};
    {
        const float* sr   = sc + l16 * SS + half * 2;   // A-fragment base (P rows)
        const float* vcol = vb + nt * 16 + l16;
        const int kc0 = khalf * (SS / 8);               // SS/4 chunks total, half each
        const int kc1 = kc0 + (SS / 8);
        for (int kc = kc0; kc < kc1; kc += 2) {
            v2f ap0 = *(const v2f*)(sr + kc * 4);       // ds_load_b64
            v2f ap1 = *(const v2f*)(sr + kc * 4 + 4);
            const int krow0 = kc * 4 + half * 2;
            v2f bp0, bp1;
            bp0.x = vcol[(size_t)krow0 * DD];
            bp0.y = vcol[(size_t)(krow0 + 1) * DD];
            bp1.x = vcol[(size_t)(krow0 + 4) * DD];
            bp1.y = vcol[(size_t)(krow0 + 5) * DD];
            oacc0 = __builtin_amdgcn_wmma_f32_16x16x4_f32(
                false, ap0, false, bp0, (short)0, oacc0, false, false);
            oacc1 = __builtin_amdgcn_wmma_f32_16x16x4_f32(
                false, ap1, false, bp1, (short)0, oacc1, false, false);
        }
    }
    v8f oacc = oacc0 + oacc1;
    // cross-wave reduction of K halves + row scaling by 1/sum
    {
        float* rw = red + wave * 256;
        #pragma unroll
        for (int j = 0; j < 8; ++j) rw[j * 32 + lane] = oacc[j];
    }
    __syncthreads();
    if (wave < 4) {
        const float* ro = red + (wave + 4) * 256;
        float* obc = ob + (half * 8) * DD + nt * 16 + l16;
        #pragma unroll
        for (int j = 0; j < 8; ++j) {
            float vsum = oacc[j] + ro[j * 32 + lane];
            __builtin_nontemporal_store(vsum * rinv[half * 8 + j], obc + j * DD);
        }
    }
}

extern "C" void kernel_launch(void* const* d_in, const int* in_sizes, int n_in,
                              void* d_out, int out_size, void* d_ws, size_t ws_size,
                              hipStream_t stream) {
    const float* q    = (const float*)d_in[0];
    const float* k    = (const float*)d_in[1];
    const float* v    = (const float*)d_in[2];
    const int*   mask = (const int*)d_in[3];

    float* out    = (float*)d_out;                          // (B,H,S,D)
    float* p_attn = out + (size_t)BB * HH * SS * DD;        // (B,H,S,S)

    (void)in_sizes; (void)n_in; (void)out_size; (void)d_ws; (void)ws_size;

    hipFuncSetAttribute((const void*)attn_fused_kernel,
                        hipFuncAttributeMaxDynamicSharedMemorySize, SMEM_BYTES);

    const int grid = BB * HH * (SS / QT);                   // 8192 blocks
    attn_fused_kernel<<<grid, NTHREADS, SMEM_BYTES, stream>>>(q, k, v, mask, out, p_attn);
}